// VIMamba_43516608643154
// MI455X (gfx1250) — compile-verified
//
#include <hip/hip_runtime.h>
#include <hip/hip_bf16.h>

// ---------------------------------------------------------------------------
// VIMamba pipeline for MI455X (gfx1250), fp32 end-to-end.
// Matrix work: V_WMMA_F32_16X16X4_F32 fed from double-buffered LDS panels
// filled with GLOBAL_LOAD_ASYNC_TO_LDS_B128 (ASYNCcnt). Scans are VALU.
// ---------------------------------------------------------------------------

typedef float v2f __attribute__((ext_vector_type(2)));
typedef float v8f __attribute__((ext_vector_type(8)));

#define BB   8
#define CC   128
#define DM   1024
#define DI   2048
#define NST  16
#define BC   (BB * CC)      // 1024 rows

#define AKP  36             // A panel row stride (floats): 144B, 16B aligned
#define BKP  36             // B panel row stride (NT)
#define BNP  80             // B panel row stride (NN): 320B, lane halves 32 banks apart
#define KB   32             // K panel depth

__device__ __forceinline__ void async_ldsb128(unsigned lds_addr, const float* gptr)
{
    asm volatile("global_load_async_to_lds_b128 %0, %1, off"
                 :: "v"(lds_addr), "v"(gptr) : "memory");
}
__device__ __forceinline__ void wait_asynccnt0()
{
    asm volatile("s_wait_asynccnt 0" ::: "memory");
}

// ---------------------------------------------------------------------------
// f32 WMMA GEMM with async-LDS double buffering.
//   B_IS_NT = true : C[m,n] = sum_k A[m*lda+k] * B[n*ldb+k]      (NT)
//   B_IS_NT = false: C[m,n] = sum_k A[m*lda+k] * B[k*ldb+n]      (NN)
// Block = 128 threads = 4 waves; block tile 64x64; wave tile 16x64.
// K is consumed in panels of 32.
// ---------------------------------------------------------------------------
template <bool B_IS_NT>
__global__ void __launch_bounds__(128)
wmma_gemm_f32(const float* __restrict__ A,
              const float* __restrict__ Bm,
              float* __restrict__ Cm,
              int M, int N, int Kd,
              int lda, int ldb, int ldc,
              long batchA, long batchB, long batchC)
{
    __shared__ float sA[2][64 * AKP];
    __shared__ float sB[2][2560];      // max(64*BKP=2304, 32*BNP=2560)

    const int t    = threadIdx.x;      // 0..127
    const int lane = t & 31;
    const int wave = t >> 5;
    const int hf   = lane >> 4;        // lane half
    const int lm   = lane & 15;
    const int mBlk = blockIdx.x * 64;
    const int nBlk = blockIdx.y * 64;

    const float* Ab = A  + (long)blockIdx.z * batchA;
    const float* Bb = Bm + (long)blockIdx.z * batchB;
    float*       Cb = Cm + (long)blockIdx.z * batchC;

    // stage one K-panel (A: 64x32, B: 64x32 NT / 32x64 NN) into LDS buffer `buf`
    auto loadPanels = [&](int buf, int k0) {
        #pragma unroll
        for (int q = 0; q < 4; ++q) {                    // A: 512 chunks? no: 64*8=512/128=4
            const int e   = q * 128 + t;                 // 16B chunk id
            const int row = e >> 3, c = e & 7;           // 8 chunks per 32-float row
            const float* g = Ab + (long)(mBlk + row) * lda + k0 + c * 4;
            const unsigned l = (unsigned)(unsigned long long)&sA[buf][row * AKP + c * 4];
            async_ldsb128(l, g);
        }
        if (B_IS_NT) {
            #pragma unroll
            for (int q = 0; q < 4; ++q) {
                const int e   = q * 128 + t;
                const int row = e >> 3, c = e & 7;       // rows are N rows
                const float* g = Bb + (long)(nBlk + row) * ldb + k0 + c * 4;
                const unsigned l = (unsigned)(unsigned long long)&sB[buf][row * BKP + c * 4];
                async_ldsb128(l, g);
            }
        } else {
            #pragma unroll
            for (int q = 0; q < 4; ++q) {
                const int e   = q * 128 + t;
                const int row = e >> 4, c = e & 15;      // rows are K rows, 16 chunks/row
                const float* g = Bb + (long)(k0 + row) * ldb + nBlk + c * 4;
                const unsigned l = (unsigned)(unsigned long long)&sB[buf][row * BNP + c * 4];
                async_ldsb128(l, g);
            }
        }
    };

    v8f acc0 = {}, acc1 = {}, acc2 = {}, acc3 = {};

    const int nkt = Kd / KB;
    loadPanels(0, 0);
    wait_asynccnt0();
    __syncthreads();

    for (int kt = 0; kt < nkt; ++kt) {
        const int buf = kt & 1;
        if (kt + 1 < nkt) loadPanels(buf ^ 1, (kt + 1) * KB);   // overlap with compute

        const float* pA = &sA[buf][(wave * 16 + lm) * AKP];
        #pragma unroll
        for (int kk = 0; kk < KB; kk += 4) {
            const int ka = kk + hf * 2;
            v2f a; a.x = pA[ka]; a.y = pA[ka + 1];

            v2f b0, b1, b2, b3;
            if (B_IS_NT) {
                const float* pB = &sB[buf][lm * BKP + ka];
                b0.x = pB[0];        b0.y = pB[1];
                b1.x = pB[16 * BKP]; b1.y = pB[16 * BKP + 1];
                b2.x = pB[32 * BKP]; b2.y = pB[32 * BKP + 1];
                b3.x = pB[48 * BKP]; b3.y = pB[48 * BKP + 1];
            } else {
                const float* pB = &sB[buf][ka * BNP + lm];
                b0.x = pB[0];  b0.y = pB[BNP];
                b1.x = pB[16]; b1.y = pB[BNP + 16];
                b2.x = pB[32]; b2.y = pB[BNP + 32];
                b3.x = pB[48]; b3.y = pB[BNP + 48];
            }

            acc0 = __builtin_amdgcn_wmma_f32_16x16x4_f32(false, a, false, b0, (short)0, acc0, false, false);
            acc1 = __builtin_amdgcn_wmma_f32_16x16x4_f32(false, a, false, b1, (short)0, acc1, false, false);
            acc2 = __builtin_amdgcn_wmma_f32_16x16x4_f32(false, a, false, b2, (short)0, acc2, false, false);
            acc3 = __builtin_amdgcn_wmma_f32_16x16x4_f32(false, a, false, b3, (short)0, acc3, false, false);
        }

        wait_asynccnt0();      // next panel fully landed in LDS
        __syncthreads();       // all waves done reading current panel
    }

    // D layout: VGPR r -> M = r + 8*hf, N = lane%16 (+16 per tile)
    const int m0 = mBlk + wave * 16;
    #pragma unroll
    for (int r = 0; r < 8; ++r) {
        const int m = m0 + r + 8 * hf;
        float* crow = Cb + (long)m * ldc + nBlk + lm;
        crow[0]  = acc0[r];
        crow[16] = acc1[r];
        crow[32] = acc2[r];
        crow[48] = acc3[r];
    }
    (void)M; (void)N;
}

// ---------------------------------------------------------------------------
// RMSNorm over last dim (one block per row).
// ---------------------------------------------------------------------------
__global__ void rmsnorm_kernel(const float* __restrict__ x,
                               const float* __restrict__ w,
                               float* __restrict__ out, int cols)
{
    __shared__ float red[256];
    const int row = blockIdx.x;
    const float* xr = x + (long)row * cols;
    float s = 0.f;
    for (int i = threadIdx.x; i < cols; i += 256) { float v = xr[i]; s += v * v; }
    red[threadIdx.x] = s; __syncthreads();
    for (int off = 128; off > 0; off >>= 1) {
        if (threadIdx.x < off) red[threadIdx.x] += red[threadIdx.x + off];
        __syncthreads();
    }
    const float inv = rsqrtf(red[0] / (float)cols + 1e-5f);
    float* orow = out + (long)row * cols;
    for (int i = threadIdx.x; i < cols; i += 256) orow[i] = w[i] * xr[i] * inv;
}

// ---------------------------------------------------------------------------
// Depthwise causal conv (K=4) + SiLU; layout (B, C, L), conv over L.
// ---------------------------------------------------------------------------
__global__ void conv_silu_kernel(const float* __restrict__ h,
                                 const float* __restrict__ w,
                                 const float* __restrict__ bias,
                                 float* __restrict__ out, int L, long total)
{
    const long idx = (long)blockIdx.x * blockDim.x + threadIdx.x;
    if (idx >= total) return;
    const int  l  = (int)(idx % L);
    const long bc = idx / L;
    const int  c  = (int)(bc % CC);
    const float* hr = h + bc * (long)L;
    float acc = bias[c];
    #pragma unroll
    for (int k = 0; k < 4; ++k) {
        const int li = l - 3 + k;
        const float xv = (li >= 0) ? hr[li] : 0.f;
        acc = fmaf(w[c * 4 + k], xv, acc);
    }
    out[idx] = acc * (1.f / (1.f + __expf(-acc)));   // SiLU
}

// ---------------------------------------------------------------------------
// 128x128 cosine matrix for the real-part DFT along the C axis.
// ---------------------------------------------------------------------------
__global__ void cosgen_kernel(float* __restrict__ cosM)
{
    const int i = blockIdx.x * blockDim.x + threadIdx.x;  // 16384
    const int k = i >> 7, n = i & 127;
    const int p = (k * n) & 127;                          // periodic reduction
    cosM[i] = cosf(6.28318530717958647692f * (float)p / 128.f);
}

// ---------------------------------------------------------------------------
// Diagonal SSM scan: one lane per (b,c); 16 states in VGPRs; L sequential.
// ---------------------------------------------------------------------------
__global__ void ssm_scan_kernel(const float* __restrict__ x,
                                const float* __restrict__ A_log,
                                const float* __restrict__ Bp,
                                const float* __restrict__ Cp,
                                const float* __restrict__ Dp,
                                const float* __restrict__ dt,
                                float* __restrict__ y, int L)
{
    const int t = blockIdx.x * blockDim.x + threadIdx.x;  // 0..B*C-1
    const int c = t % CC;
    float dA[NST], dB[NST], Cn[NST], hs[NST];
    const float dtc = dt[c];
    #pragma unroll
    for (int n = 0; n < NST; ++n) {
        const float a = -__expf(A_log[c * NST + n]);
        dA[n] = __expf(dtc * a);
        dB[n] = dtc * Bp[c * NST + n];
        Cn[n] = Cp[c * NST + n];
        hs[n] = 0.f;
    }
    const float Dc = Dp[c];
    const float* xr = x + (long)t * L;
    float*       yr = y + (long)t * L;
    for (int l = 0; l < L; ++l) {
        const float xv = xr[l];
        float acc = 0.f;
        #pragma unroll
        for (int n = 0; n < NST; ++n) {
            hs[n] = fmaf(dA[n], hs[n], dB[n] * xv);
            acc   = fmaf(hs[n], Cn[n], acc);
        }
        yr[l] = fmaf(Dc, xv, acc);
    }
}

// ---------------------------------------------------------------------------
// y = alpha * LayerNorm(y)*w+b + base   (in place on y; one block per row)
// ---------------------------------------------------------------------------
__global__ void ln_residual_kernel(float* __restrict__ y,
                                   const float* __restrict__ base,
                                   const float* __restrict__ w,
                                   const float* __restrict__ b,
                                   const float* __restrict__ alpha, int cols)
{
    __shared__ float s1[256], s2[256];
    const int row = blockIdx.x;
    float*       yr = y    + (long)row * cols;
    const float* br = base + (long)row * cols;
    float s = 0.f, ss = 0.f;
    for (int i = threadIdx.x; i < cols; i += 256) { float v = yr[i]; s += v; ss += v * v; }
    s1[threadIdx.x] = s; s2[threadIdx.x] = ss; __syncthreads();
    for (int off = 128; off > 0; off >>= 1) {
        if (threadIdx.x < off) {
            s1[threadIdx.x] += s1[threadIdx.x + off];
            s2[threadIdx.x] += s2[threadIdx.x + off];
        }
        __syncthreads();
    }
    const float mu  = s1[0] / (float)cols;
    const float var = s2[0] / (float)cols - mu * mu;
    const float inv = rsqrtf(var + 1e-5f);
    const float al  = alpha[0];
    for (int i = threadIdx.x; i < cols; i += 256) {
        const float v = yr[i];
        yr[i] = fmaf(al, fmaf((v - mu) * inv, w[i], b[i]), br[i]);
    }
}

// ---------------------------------------------------------------------------
// gate_in[b, j*DI + d] = mean_c y_j[b,c,d]
// ---------------------------------------------------------------------------
__global__ void meanc_kernel(const float* __restrict__ yl,
                             const float* __restrict__ ys,
                             const float* __restrict__ yf,
                             float* __restrict__ gatein)
{
    const int idx = blockIdx.x * blockDim.x + threadIdx.x;  // B*DI
    if (idx >= BB * DI) return;
    const int b = idx / DI, d = idx % DI;
    float s0 = 0.f, s1 = 0.f, s2 = 0.f;
    const long base = ((long)b * CC) * DI + d;
    for (int c = 0; c < CC; ++c) {
        const long o = base + (long)c * DI;
        s0 += yl[o]; s1 += ys[o]; s2 += yf[o];
    }
    const float invC = 1.f / (float)CC;
    gatein[(long)b * (3 * DI) + 0 * DI + d] = s0 * invC;
    gatein[(long)b * (3 * DI) + 1 * DI + d] = s1 * invC;
    gatein[(long)b * (3 * DI) + 2 * DI + d] = s2 * invC;
}

// ---------------------------------------------------------------------------
// gate = softmax(gate_in @ gate_W.T + gate_b); one block per batch.
// ---------------------------------------------------------------------------
__global__ void gate_kernel(const float* __restrict__ gatein,
                            const float* __restrict__ gW,
                            const float* __restrict__ gb,
                            float* __restrict__ gate, int F)
{
    __shared__ float red0[256], red1[256], red2[256];
    const int b = blockIdx.x;
    const float* gi = gatein + (long)b * F;
    float p0 = 0.f, p1 = 0.f, p2 = 0.f;
    for (int i = threadIdx.x; i < F; i += 256) {
        const float v = gi[i];
        p0 = fmaf(v, gW[i], p0);
        p1 = fmaf(v, gW[F + i], p1);
        p2 = fmaf(v, gW[2 * F + i], p2);
    }
    red0[threadIdx.x] = p0; red1[threadIdx.x] = p1; red2[threadIdx.x] = p2;
    __syncthreads();
    for (int off = 128; off > 0; off >>= 1) {
        if (threadIdx.x < off) {
            red0[threadIdx.x] += red0[threadIdx.x + off];
            red1[threadIdx.x] += red1[threadIdx.x + off];
            red2[threadIdx.x] += red2[threadIdx.x + off];
        }
        __syncthreads();
    }
    if (threadIdx.x == 0) {
        const float l0 = red0[0] + gb[0], l1 = red1[0] + gb[1], l2 = red2[0] + gb[2];
        const float mx = fmaxf(l0, fmaxf(l1, l2));
        const float e0 = __expf(l0 - mx), e1 = __expf(l1 - mx), e2 = __expf(l2 - mx);
        const float inv = 1.f / (e0 + e1 + e2);
        gate[b * 3 + 0] = e0 * inv;
        gate[b * 3 + 1] = e1 * inv;
        gate[b * 3 + 2] = e2 * inv;
    }
}

// ---------------------------------------------------------------------------
// mix = g0*y_long + g1*y_short + g2*y_freq
// ---------------------------------------------------------------------------
__global__ void combine_kernel(const float* __restrict__ yl,
                               const float* __restrict__ ys,
                               const float* __restrict__ yf,
                               const float* __restrict__ gate,
                               float* __restrict__ mix, long total)
{
    const long idx = (long)blockIdx.x * blockDim.x + threadIdx.x;
    if (idx >= total) return;
    const int b = (int)(idx / ((long)CC * DI));
    const float g0 = gate[b * 3 + 0], g1 = gate[b * 3 + 1], g2 = gate[b * 3 + 2];
    mix[idx] = g0 * yl[idx] + g1 * ys[idx] + g2 * yf[idx];
}

// ---------------------------------------------------------------------------
// out = LayerNorm(proj + residual) * w + b
// ---------------------------------------------------------------------------
__global__ void final_ln_kernel(const float* __restrict__ proj,
                                const float* __restrict__ resid,
                                const float* __restrict__ w,
                                const float* __restrict__ b,
                                float* __restrict__ out, int cols)
{
    __shared__ float s1[256], s2[256];
    const int row = blockIdx.x;
    const float* pr = proj  + (long)row * cols;
    const float* rr = resid + (long)row * cols;
    float s = 0.f, ss = 0.f;
    for (int i = threadIdx.x; i < cols; i += 256) {
        const float v = pr[i] + rr[i];
        s += v; ss += v * v;
    }
    s1[threadIdx.x] = s; s2[threadIdx.x] = ss; __syncthreads();
    for (int off = 128; off > 0; off >>= 1) {
        if (threadIdx.x < off) {
            s1[threadIdx.x] += s1[threadIdx.x + off];
            s2[threadIdx.x] += s2[threadIdx.x + off];
        }
        __syncthreads();
    }
    const float mu  = s1[0] / (float)cols;
    const float var = s2[0] / (float)cols - mu * mu;
    const float inv = rsqrtf(var + 1e-5f);
    float* orow = out + (long)row * cols;
    for (int i = threadIdx.x; i < cols; i += 256) {
        const float v = pr[i] + rr[i];
        orow[i] = fmaf((v - mu) * inv, w[i], b[i]);
    }
}

// ---------------------------------------------------------------------------
extern "C" void kernel_launch(void* const* d_in, const int* in_sizes, int n_in,
                              void* d_out, int out_size, void* d_ws, size_t ws_size,
                              hipStream_t stream)
{
    (void)in_sizes; (void)n_in; (void)out_size; (void)ws_size;

    const float* x         = (const float*)d_in[0];
    const float* norm_in_w = (const float*)d_in[1];
    const float* in_projW  = (const float*)d_in[2];   // (DI, DM) row-major
    const float* conv_w    = (const float*)d_in[3];   // (C, 4)
    const float* conv_b    = (const float*)d_in[4];
    // branches: A_log, B, C, D, dt
    const float* lA = (const float*)d_in[5];  const float* lB = (const float*)d_in[6];
    const float* lC = (const float*)d_in[7];  const float* lD = (const float*)d_in[8];
    const float* ldt = (const float*)d_in[9];
    const float* sA = (const float*)d_in[10]; const float* sB = (const float*)d_in[11];
    const float* sC = (const float*)d_in[12]; const float* sD = (const float*)d_in[13];
    const float* sdt = (const float*)d_in[14];
    const float* fA = (const float*)d_in[15]; const float* fB = (const float*)d_in[16];
    const float* fC = (const float*)d_in[17]; const float* fD = (const float*)d_in[18];
    const float* fdt = (const float*)d_in[19];
    const float* nlw = (const float*)d_in[20]; const float* nlb = (const float*)d_in[21];
    const float* nsw = (const float*)d_in[22]; const float* nsb = (const float*)d_in[23];
    const float* nfw = (const float*)d_in[24]; const float* nfb = (const float*)d_in[25];
    const float* aL  = (const float*)d_in[26];
    const float* aS  = (const float*)d_in[27];
    const float* aF  = (const float*)d_in[28];
    const float* gW  = (const float*)d_in[29]; const float* gb = (const float*)d_in[30];
    const float* out_projW = (const float*)d_in[31];  // (DM, DI) row-major
    const float* now = (const float*)d_in[32]; const float* nob = (const float*)d_in[33];

    float* ws = (float*)d_ws;
    // workspace layout (floats), with reuse: mix<-hproj, proj<-h
    float* w_h      = ws;                             // BC*DM   = 1M
    float* w_hproj  = w_h      + (long)BC * DM;       // BC*DI   = 2M
    float* w_xconv  = w_hproj  + (long)BC * DI;       // 2M
    float* w_xfreq  = w_xconv  + (long)BC * DI;       // 2M
    float* w_ylong  = w_xfreq  + (long)BC * DI;       // 2M
    float* w_yshort = w_ylong  + (long)BC * DI;       // 2M
    float* w_yfreq  = w_yshort + (long)BC * DI;       // 2M
    float* w_cos    = w_yfreq  + (long)BC * DI;       // 16384
    float* w_gatein = w_cos    + 16384;               // B*3*DI = 49152
    float* w_gate   = w_gatein + (long)BB * 3 * DI;   // 24
    float* w_mix    = w_hproj;                        // reuse
    float* w_proj   = w_h;                            // reuse

    const long totI = (long)BC * DI;   // 2M elements

    // 1) RMSNorm
    rmsnorm_kernel<<<BC, 256, 0, stream>>>(x, norm_in_w, w_h, DM);

    // 2) in_proj GEMM (NT): h(BCxDM) @ in_projW(DIxDM)^T -> hproj(BCxDI)
    wmma_gemm_f32<true><<<dim3(BC / 64, DI / 64, 1), 128, 0, stream>>>(
        w_h, in_projW, w_hproj, BC, DI, DM, DM, DM, DI, 0, 0, 0);

    // 3) causal depthwise conv + SiLU
    conv_silu_kernel<<<(int)((totI + 255) / 256), 256, 0, stream>>>(
        w_hproj, conv_w, conv_b, w_xconv, DI, totI);

    // 4) cosine matrix
    cosgen_kernel<<<64, 256, 0, stream>>>(w_cos);

    // 5) real-DFT over C axis: per-batch NN GEMM  cos(128x128) @ xconv_b(128xDI)
    wmma_gemm_f32<false><<<dim3(CC / 64, DI / 64, BB), 128, 0, stream>>>(
        w_cos, w_xconv, w_xfreq, CC, DI, CC, CC, DI, DI,
        0, (long)CC * DI, (long)CC * DI);

    // 6) three SSM scans (one lane per (b,c))
    ssm_scan_kernel<<<BC / 256, 256, 0, stream>>>(w_xconv, lA, lB, lC, lD, ldt, w_ylong, DI);
    ssm_scan_kernel<<<BC / 256, 256, 0, stream>>>(w_xconv, sA, sB, sC, sD, sdt, w_yshort, DI);
    ssm_scan_kernel<<<BC / 256, 256, 0, stream>>>(w_xfreq, fA, fB, fC, fD, fdt, w_yfreq, DI);

    // 7) per-branch LayerNorm * alpha + residual (in place)
    ln_residual_kernel<<<BC, 256, 0, stream>>>(w_ylong,  w_xconv, nlw, nlb, aL, DI);
    ln_residual_kernel<<<BC, 256, 0, stream>>>(w_yshort, w_xconv, nsw, nsb, aS, DI);
    ln_residual_kernel<<<BC, 256, 0, stream>>>(w_yfreq,  w_xfreq, nfw, nfb, aF, DI);

    // 8) gate input means over C
    meanc_kernel<<<(BB * DI + 255) / 256, 256, 0, stream>>>(w_ylong, w_yshort, w_yfreq, w_gatein);

    // 9) gate softmax
    gate_kernel<<<BB, 256, 0, stream>>>(w_gatein, gW, gb, w_gate, 3 * DI);

    // 10) weighted branch mix
    combine_kernel<<<(int)((totI + 255) / 256), 256, 0, stream>>>(
        w_ylong, w_yshort, w_yfreq, w_gate, w_mix, totI);

    // 11) out_proj GEMM (NT): mix(BCxDI) @ out_projW(DMxDI)^T -> proj(BCxDM)
    wmma_gemm_f32<true><<<dim3(BC / 64, DM / 64, 1), 128, 0, stream>>>(
        w_mix, out_projW, w_proj, BC, DM, DI, DI, DI, DM, 0, 0, 0);

    // 12) final residual + LayerNorm -> d_out
    final_ln_kernel<<<BC, 256, 0, stream>>>(w_proj, x, now, nob, (float*)d_out, DM);
}